// Net_32847909880076
// MI455X (gfx1250) — compile-verified
//
#include <hip/hip_runtime.h>

// ---------------------------------------------------------------------------
// GCN pipeline for MI455X (gfx1250, wave32).
//  - degree one-hot @ W1  ==> row-gather of W1 (no GEMM needed)
//  - edge scatters with f32 global atomics (memory-bound roofline part)
//  - h2 = x1 @ W2 done with V_WMMA_F32_16X16X4_F32 (exact f32 matrix core)
//  - fused finalize+pool with sorted-segment run accumulation (few atomics)
// ---------------------------------------------------------------------------

typedef float v2f __attribute__((ext_vector_type(2)));
typedef float v8f __attribute__((ext_vector_type(8)));

#define NNODES 524288
#define NEDGES 4194304
#define NGRAPH 64

// ---------------- degree counting (out-degree for features, in-degree for norm)
__global__ void __launch_bounds__(256)
deg_kernel(const int* __restrict__ row, const int* __restrict__ col,
           unsigned* __restrict__ deg_out, unsigned* __restrict__ deg_in, int E)
{
    int e = blockIdx.x * blockDim.x + threadIdx.x;
    if (e >= E) return;
    atomicAdd(&deg_out[row[e]], 1u);
    atomicAdd(&deg_in[col[e]], 1u);
}

// ---------------- h1[n] = W1[min(deg_out,63)],  dinv[n] = rsqrt(deg_in+1)
__global__ void __launch_bounds__(256)
node_init_kernel(const unsigned* __restrict__ deg_out, const unsigned* __restrict__ deg_in,
                 const float* __restrict__ W1, float* __restrict__ h1,
                 float* __restrict__ dinv, int N)
{
    int n = blockIdx.x * blockDim.x + threadIdx.x;
    if (n >= N) return;
    unsigned d = deg_out[n]; if (d > 63u) d = 63u;
    dinv[n] = rsqrtf((float)deg_in[n] + 1.0f);
    const float4* src = (const float4*)(W1 + (size_t)d * 16);
    float4* dst = (float4*)(h1 + (size_t)n * 16);
#pragma unroll
    for (int q = 0; q < 4; ++q) dst[q] = src[q];
}

// ---------------- layer-1 edge scatter: agg1[col] += h1[row] * dinv[r]*dinv[c]
__global__ void __launch_bounds__(256)
scatter16_kernel(const int* __restrict__ row, const int* __restrict__ col,
                 const float* __restrict__ dinv, const float* __restrict__ h1,
                 float* __restrict__ agg1, int E)
{
    int e = blockIdx.x * blockDim.x + threadIdx.x;
    if (e >= E) return;
    int r = row[e], c = col[e];
    float coef = dinv[r] * dinv[c];
    const float4* hp = (const float4*)(h1 + (size_t)r * 16);
    float* ap = agg1 + (size_t)c * 16;
#pragma unroll
    for (int q = 0; q < 4; ++q) {
        float4 h = hp[q];
        atomicAdd(ap + q * 4 + 0, h.x * coef);
        atomicAdd(ap + q * 4 + 1, h.y * coef);
        atomicAdd(ap + q * 4 + 2, h.z * coef);
        atomicAdd(ap + q * 4 + 3, h.w * coef);
    }
}

// ---------------- x1 = relu(agg1 + h1*dinv^2 + b1)   (in place in agg1)
__global__ void __launch_bounds__(256)
finalize16_kernel(float* __restrict__ agg1, const float* __restrict__ h1,
                  const float* __restrict__ dinv, const float* __restrict__ b1, int N)
{
    int idx = blockIdx.x * blockDim.x + threadIdx.x;   // one float4 per thread
    if (idx >= N * 4) return;
    int n = idx >> 2, q = idx & 3;
    float inv = dinv[n]; float i2 = inv * inv;
    float4 a = ((const float4*)agg1)[idx];
    float4 h = ((const float4*)h1)[idx];
    float4 b = ((const float4*)b1)[q];
    float4 v;
    v.x = fmaxf(a.x + h.x * i2 + b.x, 0.f);
    v.y = fmaxf(a.y + h.y * i2 + b.y, 0.f);
    v.z = fmaxf(a.z + h.z * i2 + b.z, 0.f);
    v.w = fmaxf(a.w + h.w * i2 + b.w, 0.f);
    ((float4*)agg1)[idx] = v;
}

// ---------------- h2 = x1 @ W2  via V_WMMA_F32_16X16X4_F32 (exact f32)
// One wave handles a 16-node tile. A = x1 tile (16x16), B = W2 (16x64).
// K=16 covered by 4 chained K=4 WMMAs; 4 output column blocks of 16.
// f32 A 16x4 layout (ISA 7.12.2): lanes 0-15 -> K={0,1}, lanes 16-31 -> K={2,3}.
// C/D layout: vgpr v, lane L -> M = v + 8*(L>>4), N = L&15.
__global__ void __launch_bounds__(256)
gemm_wmma_kernel(const float* __restrict__ x1, const float* __restrict__ W2,
                 float* __restrict__ h2, int nTiles)
{
    const int lane = threadIdx.x & 31;
    const int half = lane >> 4;          // 0 / 1
    const int l16  = lane & 15;
    const int wavesPerBlock = blockDim.x >> 5;
    const int wave = blockIdx.x * wavesPerBlock + (threadIdx.x >> 5);
    const int nWaves = gridDim.x * wavesPerBlock;

    // Preload B fragments once per wave: bfrag[jblock][kstep]
    v2f bfrag[4][4];
#pragma unroll
    for (int j = 0; j < 4; ++j) {
#pragma unroll
        for (int kk = 0; kk < 4; ++kk) {
            int krow = kk * 4 + half * 2;
            bfrag[j][kk].x = W2[(krow + 0) * 64 + j * 16 + l16];
            bfrag[j][kk].y = W2[(krow + 1) * 64 + j * 16 + l16];
        }
    }

    for (int t = wave; t < nTiles; t += nWaves) {
        const float* xrow = x1 + (size_t)(t * 16 + l16) * 16;
        v2f afrag[4];
#pragma unroll
        for (int kk = 0; kk < 4; ++kk) {
            int kbase = kk * 4 + half * 2;
            afrag[kk].x = xrow[kbase + 0];
            afrag[kk].y = xrow[kbase + 1];
        }
#pragma unroll
        for (int j = 0; j < 4; ++j) {
            v8f c = {};
#pragma unroll
            for (int kk = 0; kk < 4; ++kk) {
                c = __builtin_amdgcn_wmma_f32_16x16x4_f32(
                        /*neg_a=*/false, afrag[kk],
                        /*neg_b=*/false, bfrag[j][kk],
                        /*c_mod=*/(short)0, c,
                        /*reuse_a=*/false, /*reuse_b=*/false);
            }
#pragma unroll
            for (int v = 0; v < 8; ++v) {
                h2[(size_t)(t * 16 + v + 8 * half) * 64 + j * 16 + l16] = c[v];
            }
        }
    }
}

// ---------------- layer-2 edge scatter: agg2[col] += h2[row] * coef (64 feats)
// 4 threads per edge, 16 features each.
__global__ void __launch_bounds__(256)
scatter64_kernel(const int* __restrict__ row, const int* __restrict__ col,
                 const float* __restrict__ dinv, const float* __restrict__ h2,
                 float* __restrict__ agg2, int E)
{
    int idx = blockIdx.x * blockDim.x + threadIdx.x;
    if (idx >= E * 4) return;
    int e = idx >> 2, part = idx & 3;
    int r = row[e], c = col[e];
    float coef = dinv[r] * dinv[c];
    const float4* hp = (const float4*)(h2 + (size_t)r * 64 + part * 16);
    float* ap = agg2 + (size_t)c * 64 + part * 16;
#pragma unroll
    for (int q = 0; q < 4; ++q) {
        float4 h = hp[q];
        atomicAdd(ap + q * 4 + 0, h.x * coef);
        atomicAdd(ap + q * 4 + 1, h.y * coef);
        atomicAdd(ap + q * 4 + 2, h.z * coef);
        atomicAdd(ap + q * 4 + 3, h.w * coef);
    }
}

// ---------------- fused finalize2 + per-graph sum pool.
// seg is sorted; each block walks a contiguous node range keeping a running
// per-feature sum, flushing with a global atomic only on segment change.
#define NODES_PER_POOL_BLOCK 512
__global__ void __launch_bounds__(64)
pool_kernel(const float* __restrict__ agg2, const float* __restrict__ h2,
            const float* __restrict__ dinv, const float* __restrict__ b2,
            const int* __restrict__ seg, float* __restrict__ pooled, int N)
{
    int k = threadIdx.x;                       // feature 0..63
    int n0 = blockIdx.x * NODES_PER_POOL_BLOCK;
    if (n0 >= N) return;
    int n1 = n0 + NODES_PER_POOL_BLOCK; if (n1 > N) n1 = N;
    float bk = b2[k];
    float acc = 0.f;
    int cur = seg[n0];
    for (int n = n0; n < n1; ++n) {
        int s = seg[n];
        if (s != cur) {
            atomicAdd(&pooled[(size_t)cur * 64 + k], acc);
            acc = 0.f; cur = s;
        }
        float inv = dinv[n];
        acc += agg2[(size_t)n * 64 + k] + h2[(size_t)n * 64 + k] * (inv * inv) + bk;
    }
    atomicAdd(&pooled[(size_t)cur * 64 + k], acc);
}

// ---------------- tiny MLP: [64,64] -> 16 -> 16 -> 1, one thread per graph
__global__ void __launch_bounds__(64)
mlp_kernel(const float* __restrict__ pooled,
           const float* __restrict__ Wm1, const float* __restrict__ bm1,
           const float* __restrict__ Wm2, const float* __restrict__ bm2,
           const float* __restrict__ Wm3, const float* __restrict__ bm3,
           float* __restrict__ out)
{
    int b = threadIdx.x;                       // 0..63
    const float* e = pooled + (size_t)b * 64;
    float h[16];
#pragma unroll
    for (int j = 0; j < 16; ++j) {
        float s = bm1[j];
        for (int i = 0; i < 64; ++i) s += e[i] * Wm1[i * 16 + j];
        h[j] = fmaxf(s, 0.f);
    }
    float g[16];
#pragma unroll
    for (int j = 0; j < 16; ++j) {
        float s = bm2[j];
#pragma unroll
        for (int i = 0; i < 16; ++i) s += h[i] * Wm2[i * 16 + j];
        g[j] = fmaxf(s, 0.f);
    }
    float s = bm3[0];
#pragma unroll
    for (int i = 0; i < 16; ++i) s += g[i] * Wm3[i];
    out[b] = s;
}

// ---------------------------------------------------------------------------
extern "C" void kernel_launch(void* const* d_in, const int* in_sizes, int n_in,
                              void* d_out, int out_size, void* d_ws, size_t ws_size,
                              hipStream_t stream)
{
    const int E = in_sizes[0] / 2;      // 4194304
    const int N = in_sizes[2];          // 524288

    const int*   ei[2]  = { (const int*)d_in[0], (const int*)d_in[1] };
    const int*   seg[2] = { (const int*)d_in[2], (const int*)d_in[3] };
    const float* W1  = (const float*)d_in[4];
    const float* b1  = (const float*)d_in[5];
    const float* W2  = (const float*)d_in[6];
    const float* b2  = (const float*)d_in[7];
    const float* Wm1 = (const float*)d_in[8];
    const float* bm1 = (const float*)d_in[9];
    const float* Wm2 = (const float*)d_in[10];
    const float* bm2 = (const float*)d_in[11];
    const float* Wm3 = (const float*)d_in[12];
    const float* bm3 = (const float*)d_in[13];
    float* out = (float*)d_out;

    // ---- workspace carve-up (~326 MB)
    char* ws = (char*)d_ws;
    size_t off = 0;
    auto carve = [&](size_t bytes) -> void* {
        void* p = ws + off;
        off += (bytes + 255) & ~(size_t)255;
        return p;
    };
    unsigned* deg_out = (unsigned*)carve((size_t)N * 4);
    unsigned* deg_in  = (unsigned*)carve((size_t)N * 4);
    float*    dinv    = (float*)   carve((size_t)N * 4);
    float*    h1      = (float*)   carve((size_t)N * 16 * 4);
    float*    agg1    = (float*)   carve((size_t)N * 16 * 4);   // becomes x1 in place
    float*    h2      = (float*)   carve((size_t)N * 64 * 4);
    float*    agg2    = (float*)   carve((size_t)N * 64 * 4);
    float*    pooled  = (float*)   carve((size_t)NGRAPH * 64 * 4);

    hipMemsetAsync(pooled, 0, (size_t)NGRAPH * 64 * 4, stream);

    for (int gset = 0; gset < 2; ++gset) {
        const int* row = ei[gset];
        const int* col = ei[gset] + E;

        hipMemsetAsync(deg_out, 0, (size_t)N * 4, stream);
        hipMemsetAsync(deg_in,  0, (size_t)N * 4, stream);
        deg_kernel<<<(E + 255) / 256, 256, 0, stream>>>(row, col, deg_out, deg_in, E);

        node_init_kernel<<<(N + 255) / 256, 256, 0, stream>>>(deg_out, deg_in, W1, h1, dinv, N);

        hipMemsetAsync(agg1, 0, (size_t)N * 16 * 4, stream);
        scatter16_kernel<<<(E + 255) / 256, 256, 0, stream>>>(row, col, dinv, h1, agg1, E);
        finalize16_kernel<<<(N * 4 + 255) / 256, 256, 0, stream>>>(agg1, h1, dinv, b1, N);

        gemm_wmma_kernel<<<512, 256, 0, stream>>>(agg1 /*x1*/, W2, h2, N / 16);

        hipMemsetAsync(agg2, 0, (size_t)N * 64 * 4, stream);
        scatter64_kernel<<<(E * 4 + 255) / 256, 256, 0, stream>>>(row, col, dinv, h2, agg2, E);

        pool_kernel<<<(N + NODES_PER_POOL_BLOCK - 1) / NODES_PER_POOL_BLOCK, 64, 0, stream>>>(
            agg2, h2, dinv, b2, seg[gset], pooled, N);
    }

    mlp_kernel<<<1, 64, 0, stream>>>(pooled, Wm1, bm1, Wm2, bm2, Wm3, bm3, out);
    (void)n_in; (void)out_size; (void)ws_size;
}